// S4Recurrent_20401094656327
// MI455X (gfx1250) — compile-verified
//
#include <hip/hip_runtime.h>
#include <math.h>

#define SEQ    4096
#define NBATCH 4
#define DMODEL 1024
#define NS     64          // state dim

typedef __attribute__((ext_vector_type(2))) float v2f;
typedef __attribute__((ext_vector_type(8))) float v8f;

// ---------------------------------------------------------------------------
// Wave-level 64x64x64 f32 matmul tile helpers using V_WMMA_F32_16X16X4_F32.
// Layouts per CDNA5 ISA 7.12.2:
//   A (16x4 f32):  lanes 0-15 hold rows M=0..15 with K=0 (VGPR0), K=1 (VGPR1);
//                  lanes 16-31 hold K=2 (VGPR0), K=3 (VGPR1).
//   B (4x16 f32):  mirrored: lanes 0-15 cols N, K=0/1 in VGPR0/1; lanes 16-31 K=2/3.
//   C/D (16x16 f32): VGPR r -> M=r (lanes 0-15) / M=r+8 (lanes 16-31), N=lane&15.
// ---------------------------------------------------------------------------
__device__ __forceinline__ v8f wave_mm64(const float* X, const float* Y,
                                         int mi, int ni, float scale) {
  const int lane = threadIdx.x & 31;
  const int half = lane >> 4;
  const int l15  = lane & 15;
  const int row  = mi * 16 + l15;
  const int col  = ni * 16 + l15;
  v8f acc = {0.f, 0.f, 0.f, 0.f, 0.f, 0.f, 0.f, 0.f};
#pragma unroll
  for (int kk = 0; kk < 16; ++kk) {
    const int kb = kk * 4 + half * 2;
    v2f a, b;
    a.x = X[row * 64 + kb];
    a.y = X[row * 64 + kb + 1];
    b.x = Y[kb * 64 + col];
    b.y = Y[(kb + 1) * 64 + col];
    acc = __builtin_amdgcn_wmma_f32_16x16x4_f32(false, a, false, b,
                                                (short)0, acc, false, false);
  }
#pragma unroll
  for (int r = 0; r < 8; ++r) acc[r] *= scale;
  return acc;
}

__device__ __forceinline__ void wave_store64(float* D, int mi, int ni, v8f acc) {
  const int lane = threadIdx.x & 31;
  const int rof  = (lane >> 4) * 8;
  const int col  = ni * 16 + (lane & 15);
#pragma unroll
  for (int r = 0; r < 8; ++r) D[(mi * 16 + rof + r) * 64 + col] = acc[r];
}

__device__ __forceinline__ v8f wave_load64(const float* S, int mi, int ni) {
  const int lane = threadIdx.x & 31;
  const int rof  = (lane >> 4) * 8;
  const int col  = ni * 16 + (lane & 15);
  v8f r;
#pragma unroll
  for (int q = 0; q < 8; ++q) r[q] = S[(mi * 16 + rof + q) * 64 + col];
  return r;
}

__device__ __forceinline__ v8f wave_itile(int mi, int ni) {
  const int lane = threadIdx.x & 31;
  const int rof  = (lane >> 4) * 8;
  const int col  = ni * 16 + (lane & 15);
  v8f r;
#pragma unroll
  for (int q = 0; q < 8; ++q) r[q] = ((mi * 16 + rof + q) == col) ? 1.f : 0.f;
  return r;
}

// ---------------------------------------------------------------------------
// Prep kernel (1 block x 512 threads = 16 waves):
//   expm(dt*A) via scaling+squaring Taylor; B_bar = dt*phi1(dt*A)*B via the
//   doubling phi1(2M) = (E+I)phi1(M)/2; then k[64j+i] = (c A^i)·(A^64j b)
//   = (W @ V)[i][j] with W,V from short recurrences and WMMA matmuls.
// ---------------------------------------------------------------------------
__global__ __launch_bounds__(512)
void s4_prep_kernel(const float* __restrict__ A,      // 64x64 row-major
                    const float* __restrict__ Bcol,   // 64
                    const float* __restrict__ Crow,   // 64
                    const float* __restrict__ log_dt, // 1
                    float* __restrict__ k_out)        // 4096
{
  __shared__ float MM0[64 * 64];
  __shared__ float MM1[64 * 64];
  __shared__ float MM2[64 * 64];
  __shared__ float vecb[64];
  __shared__ float rowsum[64];
  __shared__ float s_scale, s_dt;
  __shared__ int   s_shift;

  const int tid = threadIdx.x;
  const int w   = tid >> 5;
  const int mi  = w >> 2;
  const int ni  = w & 3;

  // ---- scaling: s so that ||dt*A/2^s||_inf <= 0.5 ----
  if (tid < 64) {
    float s = 0.f;
    for (int j = 0; j < 64; ++j) s += fabsf(A[tid * 64 + j]);
    rowsum[tid] = s;
  }
  __syncthreads();
  if (tid == 0) {
    float dt = expf(log_dt[0]);
    float nm = 0.f;
    for (int i = 0; i < 64; ++i) nm = fmaxf(nm, rowsum[i]);
    nm *= dt;
    int sh = 0;
    while (nm > 0.5f && sh < 48) { nm *= 0.5f; ++sh; }
    s_dt = dt; s_shift = sh; s_scale = dt * exp2f((float)(-sh));
  }
  __syncthreads();

  // Ms = (dt/2^s)*A into MM2, Taylor running term into MM0
  {
    const float sc = s_scale;
    for (int i = tid; i < 4096; i += 512) {
      float v = A[i] * sc;
      MM2[i] = v;
      MM0[i] = v;
    }
  }
  __syncthreads();

  // Taylor: E = sum Ms^k/k!, P = phi1 = sum Ms^k/(k+1)!  (tiles in registers)
  v8f Et  = wave_itile(mi, ni);
  v8f Pt  = Et;
  v8f MsT = wave_load64(MM2, mi, ni);
#pragma unroll
  for (int r = 0; r < 8; ++r) { Et[r] += MsT[r]; Pt[r] += 0.5f * MsT[r]; }

  float* cur = MM0;
  float* nxt = MM1;
  for (int kk = 2; kk <= 16; ++kk) {
    v8f T = wave_mm64(cur, MM2, mi, ni, 1.0f / (float)kk);   // T_k = T_{k-1}@Ms/k
    wave_store64(nxt, mi, ni, T);
    const float pf = 1.0f / (float)(kk + 1);
#pragma unroll
    for (int r = 0; r < 8; ++r) { Et[r] += T[r]; Pt[r] += pf * T[r]; }
    float* t = cur; cur = nxt; nxt = t;
    __syncthreads();
  }
  wave_store64(MM0, mi, ni, Et);   // E
  wave_store64(MM1, mi, ni, Pt);   // P = phi1
  __syncthreads();

  // Squaring: P <- 0.5*(E@P + P);  E <- E@E   (s_shift times)
  for (int it = 0; it < s_shift; ++it) {
    v8f t1 = wave_mm64(MM0, MM1, mi, ni, 0.5f);
    wave_store64(MM2, mi, ni, t1);
    __syncthreads();
    for (int i = tid; i < 4096; i += 512) MM1[i] = MM2[i] + 0.5f * MM1[i];
    __syncthreads();
    v8f t2 = wave_mm64(MM0, MM0, mi, ni, 1.0f);
    __syncthreads();                 // all reads of MM0 done before overwrite
    wave_store64(MM0, mi, ni, t2);
    __syncthreads();
  }
  // Now: MM0 = A_bar = expm(dt*A), MM1 = phi1(dt*A)

  // bvec = dt * phi1 @ B   (== A^{-1}(A_bar - I)B exactly)
  if (tid < 64) {
    float s = 0.f;
    for (int r = 0; r < 64; ++r) s += MM1[tid * 64 + r] * Bcol[r];
    vecb[tid] = s * s_dt;
  }
  __syncthreads();

  // W rows: w_i = c * A_bar^i  (i = 0..63) into MM2 (P is dead)
  if (tid < 64) MM2[tid] = Crow[tid];
  __syncthreads();
  for (int i = 1; i < 64; ++i) {
    if (tid < 64) {
      float s = 0.f;
      for (int r = 0; r < 64; ++r) s += MM2[(i - 1) * 64 + r] * MM0[r * 64 + tid];
      MM2[i * 64 + tid] = s;
    }
    __syncthreads();
  }

  // Repeated squaring chain to A_bar^64 (A_bar dead after first step):
  // A^2->MM1, A^4->MM0, A^8->MM1, A^16->MM0, A^32->MM1, A^64->MM0
  {
    v8f t = wave_mm64(MM0, MM0, mi, ni, 1.0f); __syncthreads();
    wave_store64(MM1, mi, ni, t); __syncthreads();
    t = wave_mm64(MM1, MM1, mi, ni, 1.0f); __syncthreads();
    wave_store64(MM0, mi, ni, t); __syncthreads();
    t = wave_mm64(MM0, MM0, mi, ni, 1.0f); __syncthreads();
    wave_store64(MM1, mi, ni, t); __syncthreads();
    t = wave_mm64(MM1, MM1, mi, ni, 1.0f); __syncthreads();
    wave_store64(MM0, mi, ni, t); __syncthreads();
    t = wave_mm64(MM0, MM0, mi, ni, 1.0f); __syncthreads();
    wave_store64(MM1, mi, ni, t); __syncthreads();
    t = wave_mm64(MM1, MM1, mi, ni, 1.0f); __syncthreads();
    wave_store64(MM0, mi, ni, t); __syncthreads();
  }
  // MM0 = A_bar^64

  // V cols: v_j = (A_bar^64)^j * bvec into MM1 (column-major: MM1[n*64+j])
  if (tid < 64) MM1[tid * 64 + 0] = vecb[tid];
  __syncthreads();
  for (int j = 1; j < 64; ++j) {
    if (tid < 64) {
      float s = 0.f;
      for (int r = 0; r < 64; ++r) s += MM0[tid * 64 + r] * MM1[r * 64 + (j - 1)];
      MM1[tid * 64 + j] = s;
    }
    __syncthreads();
  }

  // K = W @ V ;  k[64*j + i] = K[i][j]
  {
    v8f Kt = wave_mm64(MM2, MM1, mi, ni, 1.0f);
    const int lane = threadIdx.x & 31;
    const int rof  = (lane >> 4) * 8;
    const int col  = ni * 16 + (lane & 15);
#pragma unroll
    for (int r = 0; r < 8; ++r) {
      const int row = mi * 16 + rof + r;
      k_out[col * 64 + row] = Kt[r];
    }
  }
}

// ---------------------------------------------------------------------------
// Causal convolution: yv[b][t] = sum_{s<=t} k[t-s]*u0[b][s] + d*u0[b][t]
// grid (SEQ/256, NBATCH), 256 threads; LDS-staged chunks of 256 u / 512 k.
// ---------------------------------------------------------------------------
__global__ __launch_bounds__(256)
void s4_conv_kernel(const float* __restrict__ u,       // (B,SEQ,DMODEL)
                    const float* __restrict__ k,       // SEQ
                    const float* __restrict__ Dscalar, // 1
                    float* __restrict__ yv)            // (B,SEQ)
{
  __shared__ float su[256];
  __shared__ float sk[512];
  const int b   = blockIdx.y;
  const int t0  = blockIdx.x * 256;
  const int tid = threadIdx.x;
  const int t   = t0 + tid;

  float acc = 0.f;
  for (int s0 = 0; s0 <= t0; s0 += 256) {
    __syncthreads();
    su[tid] = u[(size_t)(b * SEQ + s0 + tid) * DMODEL];   // u0 = u[:,:,0]
    const int w0 = t0 - s0 - 255;
    for (int j = tid; j < 512; j += 256) {
      const int kk = w0 + j;
      sk[j] = (kk >= 0 && kk < SEQ) ? k[kk] : 0.f;
    }
    __syncthreads();
#pragma unroll 8
    for (int j = 0; j < 256; ++j)
      acc = fmaf(su[j], sk[tid + 255 - j], acc);          // tau = t-(s0+j)
  }
  const float ut = u[(size_t)(b * SEQ + t) * DMODEL];
  yv[b * SEQ + t] = acc + Dscalar[0] * ut;
}

// ---------------------------------------------------------------------------
// Broadcast yv[b][t] across 1024 channels: 64 MB of float4 stores (BW-bound).
// ---------------------------------------------------------------------------
__global__ __launch_bounds__(256)
void s4_bcast_kernel(const float* __restrict__ yv, float4* __restrict__ out) {
  const int row = blockIdx.x;                 // b*SEQ + t, 0..16383
  const float v = yv[row];
  const float4 vv = make_float4(v, v, v, v);
  out[(size_t)row * (DMODEL / 4) + threadIdx.x] = vv;
}

extern "C" void kernel_launch(void* const* d_in, const int* in_sizes, int n_in,
                              void* d_out, int out_size, void* d_ws, size_t ws_size,
                              hipStream_t stream) {
  (void)in_sizes; (void)n_in; (void)out_size; (void)ws_size;
  const float* u      = (const float*)d_in[0];
  const float* A      = (const float*)d_in[1];
  const float* B      = (const float*)d_in[2];
  const float* C      = (const float*)d_in[3];
  const float* D      = (const float*)d_in[4];
  const float* log_dt = (const float*)d_in[5];

  float* ws = (float*)d_ws;
  float* k  = ws;             // 4096 floats
  float* yv = ws + SEQ;       // NBATCH*SEQ floats

  s4_prep_kernel<<<1, 512, 0, stream>>>(A, B, C, log_dt, k);
  dim3 cgrid(SEQ / 256, NBATCH);
  s4_conv_kernel<<<cgrid, 256, 0, stream>>>(u, k, D, yv);
  s4_bcast_kernel<<<NBATCH * SEQ, DMODEL / 4, 0, stream>>>(yv, (float4*)d_out);
}